// FairGAT_38113539785178
// MI455X (gfx1250) — compile-verified
//
#include <hip/hip_runtime.h>
#include <hip/hip_bf16.h>

#define NN 100000
#define EE 1600000
#define ET (EE + NN)

typedef __attribute__((ext_vector_type(2))) float v2f;
typedef __attribute__((ext_vector_type(8))) float v8f;

__device__ __forceinline__ unsigned flip_f32(float f) {
    unsigned u = __float_as_uint(f);
    unsigned mask = (unsigned)(-(int)(u >> 31)) | 0x80000000u;
    return u ^ mask;
}
__device__ __forceinline__ float unflip_f32(unsigned k) {
    unsigned u = (k & 0x80000000u) ? (k ^ 0x80000000u) : ~k;
    return __uint_as_float(u);
}

// ---------------------------------------------------------------------------
// Generic fp32 WMMA GEMM: C[M x Ncols] = A[M x K] * B[K x Ncols] (+bias, relu)
// One wave computes a 16x16 C tile via V_WMMA_F32_16X16X4_F32.
// B panel staged in LDS, pre-swizzled into WMMA fragment order so the inner
// loop is: global_load_b64 (A frag) + ds_load_b64 (B frag) + v_wmma.
// No value masking in the loop: OOB rows are clamped (discarded at store),
// OOB cols read stale in-bounds LDS (their C columns are never stored).
// ---------------------------------------------------------------------------
__global__ __launch_bounds__(256) void gemm_wmma_f32(
    const float* __restrict__ A, int lda,
    const float* __restrict__ B, int Ncols, int K,
    const float* __restrict__ bias,
    float* __restrict__ C, int ldc,
    int M, int relu, int rowTilesPerBlock)
{
    // fragment-order panel: [K/4 steps][colTiles][lane 0..31][2 floats]
    __shared__ float Bs[128 * 64];
    const int tid = threadIdx.x;
    const int colTiles = (Ncols + 15) >> 4;

    // cooperative stage + swizzle of the full B panel
    // element B[k][n] -> Bs[((k>>2)*colTiles + n>>4)*64 + lane*2 + (k&1)],
    //   lane = ((k>>1)&1)*16 + (n&15)
    const int totB = K * Ncols;
    for (int i = tid; i < totB; i += 256) {
        const int k = i / Ncols;
        const int n = i - k * Ncols;
        const int lane = (((k >> 1) & 1) << 4) + (n & 15);
        const int idx = (((k >> 2) * colTiles + (n >> 4)) << 6) + (lane << 1) + (k & 1);
        Bs[idx] = B[i];
    }
    __syncthreads();

    const int wave = tid >> 5;
    const int lane = tid & 31;
    const int rt = wave / colTiles;
    const int ct = wave % colTiles;
    if (rt >= rowTilesPerBlock) return;  // wave-uniform exit (EXEC stays full)

    const long rowBase = (long)blockIdx.x * (16 * rowTilesPerBlock) + rt * 16;
    const int colBase = ct * 16;

    // A row this lane loads (clamped for tail blocks: garbage rows never stored)
    long mrow = rowBase + (lane & 15);
    if (mrow > M - 1) mrow = M - 1;
    const int khalf = (lane >> 4) * 2;  // lanes 16..31 hold K+2,K+3

    const v2f* __restrict__ Ap = (const v2f*)(A + mrow * lda + khalf);
    const float* bp = Bs + (ct << 6) + (lane << 1);
    const int bstride = colTiles << 6;
    const int steps = K >> 2;
    __builtin_prefetch(Ap, 0, 0);

    v8f acc = {};
    #pragma unroll 4
    for (int s = 0; s < steps; ++s) {
        const v2f a = Ap[2 * s];                        // global_load_b64
        const v2f b = *(const v2f*)(bp + s * bstride);  // ds_load_b64 (conflict-free)
        acc = __builtin_amdgcn_wmma_f32_16x16x4_f32(
            false, a, false, b, (short)0, acc, false, false);
    }

    const int ncol = colBase + (lane & 15);
    if (ncol < Ncols) {
        const float bv = bias ? bias[ncol] : 0.0f;
        #pragma unroll
        for (int v = 0; v < 8; ++v) {
            const long r = rowBase + v + 8 * (lane >> 4);
            if (r < M) {
                float val = acc[v] + bv;
                if (relu) val = val > 0.0f ? val : 0.0f;
                C[r * ldc + ncol] = val;
            }
        }
    }
}

// ---------------------------------------------------------------------------
// small utility kernels
// ---------------------------------------------------------------------------
__global__ __launch_bounds__(256) void fill_f32(float* p, float v, int n) {
    int i = blockIdx.x * 256 + threadIdx.x;
    if (i < n) p[i] = v;
}
__global__ __launch_bounds__(256) void fill_u32(unsigned* p, unsigned v, int n) {
    int i = blockIdx.x * 256 + threadIdx.x;
    if (i < n) p[i] = v;
}
__global__ __launch_bounds__(256) void init_agg_bias(float* agg, const float* __restrict__ b, int total) {
    int i = blockIdx.x * 256 + threadIdx.x;
    if (i < total) agg[i] = b[i & 63];
}
__global__ __launch_bounds__(256) void copy_fair(float* xbuf, const float* __restrict__ fair, int total) {
    int i = blockIdx.x * 256 + threadIdx.x;
    if (i < total) {
        long n = i >> 6;
        int c = i & 63;
        xbuf[n * 128 + 64 + c] = fair[i];
    }
}

// per-(node, head) attention coefficients
__global__ __launch_bounds__(256) void alpha_kernel(
    const float* __restrict__ h, const float* __restrict__ a_s,
    const float* __restrict__ a_d, float* asrc, float* adst, int total)
{
    int i = blockIdx.x * 256 + threadIdx.x;
    if (i >= total) return;
    const int head = i & 1;
    const long node = i >> 1;
    const float* hp = h + node * 64 + head * 32;
    const float* sv = a_s + head * 32;
    const float* dv = a_d + head * 32;
    float s = 0.0f, d = 0.0f;
    #pragma unroll
    for (int c = 0; c < 32; ++c) {
        const float hv = hp[c];
        s = fmaf(hv, sv[c], s);
        d = fmaf(hv, dv[c], d);
    }
    asrc[i] = s;
    adst[i] = d;
}

// pass 1: leaky-relu logits + segment max (monotonic-uint atomic max)
__global__ __launch_bounds__(256) void edge_logit_max(
    const int* __restrict__ ei, const float* __restrict__ asrc,
    const float* __restrict__ adst, float* __restrict__ evalb,
    unsigned* __restrict__ menc)
{
    int t = blockIdx.x * 256 + threadIdx.x;
    if (t >= ET * 2) return;
    const int e = t >> 1, head = t & 1;
    int s, d;
    if (e < EE) { s = ei[e]; d = ei[EE + e]; } else { s = d = e - EE; }
    float v = asrc[(long)s * 2 + head] + adst[(long)d * 2 + head];
    v = v > 0.0f ? v : 0.2f * v;
    evalb[t] = v;
    atomicMax(&menc[(long)d * 2 + head], flip_f32(v));
}

__global__ __launch_bounds__(256) void decode_m(const unsigned* __restrict__ menc, float* mflt, int n) {
    int i = blockIdx.x * 256 + threadIdx.x;
    if (i >= n) return;
    float f = unflip_f32(menc[i]);
    if (!__builtin_isfinite(f)) f = 0.0f;
    mflt[i] = f;
}

// pass 2: p = exp(e - m[dst]); den[dst] += p
__global__ __launch_bounds__(256) void edge_exp_den(
    const int* __restrict__ ei, float* __restrict__ evalb,
    const float* __restrict__ mflt, float* __restrict__ den)
{
    int t = blockIdx.x * 256 + threadIdx.x;
    if (t >= ET * 2) return;
    const int e = t >> 1, head = t & 1;
    const int d = (e < EE) ? ei[EE + e] : (e - EE);
    const float p = __expf(evalb[t] - mflt[(long)d * 2 + head]);
    evalb[t] = p;
    atomicAdd(&den[(long)d * 2 + head], p);
}

// pass 3: agg[dst] += (p/den) * h[src]; one thread per (edge, float4-channel-group)
__global__ __launch_bounds__(256) void edge_aggregate(
    const int* __restrict__ ei, const float* __restrict__ evalb,
    const float* __restrict__ den, const float* __restrict__ h,
    float* __restrict__ agg)
{
    int t = blockIdx.x * 256 + threadIdx.x;
    if (t >= ET * 16) return;
    const int e = t >> 4;
    const int c4 = t & 15;
    int s, d;
    if (e < EE) { s = ei[e]; d = ei[EE + e]; } else { s = d = e - EE; }
    const int head = c4 >> 3;
    const float alpha = evalb[(long)e * 2 + head] / (den[(long)d * 2 + head] + 1e-16f);
    const float4 hv = ((const float4*)(h + (long)s * 64))[c4];
    float* op = agg + (long)d * 64 + c4 * 4;
    atomicAdd(op + 0, alpha * hv.x);
    atomicAdd(op + 1, alpha * hv.y);
    atomicAdd(op + 2, alpha * hv.z);
    atomicAdd(op + 3, alpha * hv.w);
}

// batchnorm statistics: per-column partial sums -> global atomics
__global__ __launch_bounds__(256) void bn_stats(const float* __restrict__ agg, float* __restrict__ stats) {
    const int c = threadIdx.x & 63;
    const int rowsPerBlk = 256 >> 6;  // 4
    float s = 0.0f, q = 0.0f;
    for (long r = (long)blockIdx.x * rowsPerBlk + (threadIdx.x >> 6); r < NN;
         r += (long)gridDim.x * rowsPerBlk) {
        const float v = agg[r * 64 + c];
        s += v;
        q += v * v;
    }
    atomicAdd(&stats[c], s);
    atomicAdd(&stats[64 + c], q);
}

__global__ void bn_finalize(float* stats) {
    const int c = threadIdx.x;
    if (c >= 64) return;
    const float mu = stats[c] / (float)NN;
    const float var = stats[64 + c] / (float)NN - mu * mu;
    stats[128 + c] = mu;
    stats[192 + c] = rsqrtf(var + 1e-5f);
}

// normalize + scale/shift + relu, writing into ld=128 buffer (concat layout)
__global__ __launch_bounds__(256) void bn_apply(
    const float* __restrict__ agg, const float* __restrict__ stats,
    const float* __restrict__ g, const float* __restrict__ be,
    float* __restrict__ xbuf, int total)
{
    int i = blockIdx.x * 256 + threadIdx.x;
    if (i >= total) return;
    const long n = i >> 6;
    const int c = i & 63;
    float v = (agg[i] - stats[128 + c]) * stats[192 + c] * g[c] + be[c];
    v = v > 0.0f ? v : 0.0f;
    xbuf[n * 128 + c] = v;
}

// ---------------------------------------------------------------------------
// launcher
// ---------------------------------------------------------------------------
static inline int cdiv(long a, long b) { return (int)((a + b - 1) / b); }

extern "C" void kernel_launch(void* const* d_in, const int* in_sizes, int n_in,
                              void* d_out, int out_size, void* d_ws, size_t ws_size,
                              hipStream_t stream) {
    const float* x    = (const float*)d_in[0];
    const int*   ei   = (const int*)d_in[1];
    const float* fair = (const float*)d_in[2];
    const float* lw1  = (const float*)d_in[21];
    const float* lb1  = (const float*)d_in[22];
    const float* lw2  = (const float*)d_in[23];
    const float* lb2  = (const float*)d_in[24];

    // workspace layout (floats)
    float* fws   = (float*)d_ws;
    float* hbuf  = fws;                         // NN*64
    float* agg   = hbuf + (long)NN * 64;        // NN*64
    float* xbuf  = agg + (long)NN * 64;         // NN*128 (cols 0..63 = x, 64..127 = fair)
    float* asrc  = xbuf + (long)NN * 128;       // NN*2
    float* adst  = asrc + NN * 2;               // NN*2
    unsigned* menc = (unsigned*)(adst + NN * 2);// NN*2
    float* mflt  = (float*)(menc + NN * 2);     // NN*2
    float* den   = mflt + NN * 2;               // NN*2
    float* evalb = den + NN * 2;                // ET*2
    float* stats = evalb + (long)ET * 2;        // 256
    float* hmid  = stats + 256;                 // NN*32

    const int thr = 256;

    // stage the fair embedding once into concat columns 64..127
    copy_fair<<<cdiv((long)NN * 64, thr), thr, 0, stream>>>(xbuf, fair, NN * 64);

    const float* xcur = x;
    int K = 128;
    for (int l = 0; l < 3; ++l) {
        const float* W   = (const float*)d_in[3 + 6 * l + 0];
        const float* a_s = (const float*)d_in[3 + 6 * l + 1];
        const float* a_d = (const float*)d_in[3 + 6 * l + 2];
        const float* b   = (const float*)d_in[3 + 6 * l + 3];
        const float* g   = (const float*)d_in[3 + 6 * l + 4];
        const float* be  = (const float*)d_in[3 + 6 * l + 5];

        // h = xcur @ W   (Ncols=64 -> 4 col tiles, 2 row tiles per block)
        gemm_wmma_f32<<<cdiv(NN, 32), 256, 0, stream>>>(
            xcur, 128, W, 64, K, nullptr, hbuf, 64, NN, 0, 2);

        alpha_kernel<<<cdiv((long)NN * 2, thr), thr, 0, stream>>>(
            hbuf, a_s, a_d, asrc, adst, NN * 2);

        fill_u32<<<cdiv((long)NN * 2, thr), thr, 0, stream>>>(menc, 0x007FFFFFu, NN * 2); // flip(-inf)
        fill_f32<<<cdiv((long)NN * 2, thr), thr, 0, stream>>>(den, 0.0f, NN * 2);
        init_agg_bias<<<cdiv((long)NN * 64, thr), thr, 0, stream>>>(agg, b, NN * 64);

        edge_logit_max<<<cdiv((long)ET * 2, thr), thr, 0, stream>>>(ei, asrc, adst, evalb, menc);
        decode_m<<<cdiv((long)NN * 2, thr), thr, 0, stream>>>(menc, mflt, NN * 2);
        edge_exp_den<<<cdiv((long)ET * 2, thr), thr, 0, stream>>>(ei, evalb, mflt, den);
        edge_aggregate<<<cdiv((long)ET * 16, thr), thr, 0, stream>>>(ei, evalb, den, hbuf, agg);

        fill_f32<<<1, 256, 0, stream>>>(stats, 0.0f, 128);
        bn_stats<<<512, 256, 0, stream>>>(agg, stats);
        bn_finalize<<<1, 64, 0, stream>>>(stats);
        bn_apply<<<cdiv((long)NN * 64, thr), thr, 0, stream>>>(agg, stats, g, be, xbuf, NN * 64);

        xcur = xbuf;
        K = 64;
    }

    // MLP head: hmid = relu(concat(x, fair) @ lw1 + lb1)   (Ncols=32 -> 2 col tiles, 4 row tiles)
    gemm_wmma_f32<<<cdiv(NN, 64), 256, 0, stream>>>(
        xbuf, 128, lw1, 32, 128, lb1, hmid, 32, NN, 1, 4);

    // out = hmid @ lw2 + lb2   (Ncols=40 -> 3 col tiles, 2 row tiles, tail-masked)
    gemm_wmma_f32<<<cdiv(NN, 32), 256, 0, stream>>>(
        hmid, 32, lw2, 40, 32, lb2, (float*)d_out, 40, NN, 0, 2);
}